// E3Hamiltonian_87342454931494
// MI455X (gfx1250) — compile-verified
//
#include <hip/hip_runtime.h>
#include <cmath>

// ============================================================================
// E3 Hamiltonian rotation, MI455X (gfx1250, wave32)
//   Stage 0 (tiny, fp64): build e3nn CG matrices on device (d_ws).
//   Stage 1: Hz = CG * rme  as f32 WMMA GEMMs, branch-free unrolled K-loops,
//            zero-padded B fragments staged in LDS; s-* pairs are identity.
//   Stage 2: per-edge HR = D_l1 * Hz * D_l2^T on VALU, closed-form D1/D2.
// ============================================================================

typedef float v2f __attribute__((ext_vector_type(2)));
typedef float v8f __attribute__((ext_vector_type(8)));

#define EPB 64               // edges per block (Hz LDS = 64*196*4 = 50176 B)
#define NBLK 14              // (pair,l3) wigner blocks

// padded B-fragment LDS layout bases (floats)
#define CGP_PP 0             // 3 ksteps * 2 * 16 * 2 = 192
#define CGP_PD 192           // 4 * 2 * 16 * 2       = 256
#define CGP_DP 448           // 256
#define CGP_DD 704           // 7 * 2 * 32 * 2       = 896
#define CGP_TOT 1600

__device__ __forceinline__ int imax2(int a, int b) { return a > b ? a : b; }
__device__ __forceinline__ int imin2(int a, int b) { return a < b ? a : b; }

// ---------------------------------------------------------------------------
// fp64 CG construction (exact port of the reference numpy code)
// ---------------------------------------------------------------------------
__device__ double dfact(int n) { double r = 1.0; for (int i = 2; i <= n; ++i) r *= i; return r; }

__device__ double dsu2cg(int j1, int j2, int j3, int m1, int m2, int m3) {
    if (m3 != m1 + m2) return 0.0;
    int vmin = imax2(imax2(-j1 + j2 + m3, -j1 + m1), 0);
    int vmax = imin2(imin2(j2 + j3 + m1, j3 - j1 + j2), j3 + m3);
    double C = sqrt((2.0 * j3 + 1.0) * dfact(j3 + j1 - j2) * dfact(j3 - j1 + j2) * dfact(j1 + j2 - j3)
                    * dfact(j3 + m3) * dfact(j3 - m3)
                    / (dfact(j1 + j2 + j3 + 1) * dfact(j1 - m1) * dfact(j1 + m1)
                       * dfact(j2 - m2) * dfact(j2 + m2)));
    double S = 0.0;
    for (int v = vmin; v <= vmax; ++v) {
        double sgn = ((v + j2 + m2) & 1) ? -1.0 : 1.0;
        S += sgn * dfact(j2 + j3 + m1 - v) * dfact(j1 - m1 + v)
             / (dfact(v) * dfact(j3 - j1 + j2 - v) * dfact(j3 + m3 - v) * dfact(v + j1 - j2 - m3));
    }
    return C * S;
}

// Q = (-i)^l * change_basis_real_to_complex(l), split re/im, row-major (2l+1)^2
__device__ void dchange_basis(int l, double* qr, double* qi) {
    int L = 2 * l + 1;
    for (int i = 0; i < L * L; ++i) { qr[i] = 0.0; qi[i] = 0.0; }
    const double s = 0.7071067811865476;
    for (int m = -l; m < 0; ++m) {
        qr[(l + m) * L + (l - m)] = s;     // q[l+m, l+|m|] =  1/sqrt2
        qi[(l + m) * L + (l + m)] = -s;    // q[l+m, l-|m|] = -i/sqrt2
    }
    qr[l * L + l] = 1.0;
    for (int m = 1; m <= l; ++m) {
        double sgn = (m & 1) ? -1.0 : 1.0;
        qr[(l + m) * L + (l + m)] = sgn * s;
        qi[(l + m) * L + (l - m)] = sgn * s;
    }
    int ph = l & 3;                        // multiply by (-i)^l
    for (int i = 0; i < L * L; ++i) {
        double r = qr[i], im = qi[i];
        if (ph == 1)      { qr[i] =  im; qi[i] = -r;  }
        else if (ph == 2) { qr[i] = -r;  qi[i] = -im; }
        else if (ph == 3) { qr[i] = -im; qi[i] =  r;  }
    }
}

// block tables: pairs {(1,1),(1,2),(2,1),(2,2)} x l3 ranges
__device__ __constant__ int c_bl1[NBLK]  = {1,1,1, 1,1,1, 2,2,2, 2,2,2,2,2};
__device__ __constant__ int c_bl2[NBLK]  = {1,1,1, 2,2,2, 1,1,1, 2,2,2,2,2};
__device__ __constant__ int c_bl3[NBLK]  = {0,1,2, 1,2,3, 1,2,3, 0,1,2,3,4};
__device__ __constant__ int c_boff[NBLK] = {0,9,36, 81,126,201, 306,351,426, 531,556,631,756,931};
__device__ __constant__ int c_bsz[NBLK]  = {9,27,45, 45,75,105, 45,75,105, 25,75,125,175,225};
__device__ __constant__ int c_bpair[NBLK]= {0,0,0, 1,1,1, 2,2,2, 3,3,3,3,3};
__device__ __constant__ int c_brofs[NBLK]= {0,1,4, 0,3,8, 0,3,8, 0,1,4,9,16};
__device__ __constant__ int c_cgbase[4]  = {0, 81, 306, 531};
__device__ __constant__ int c_Rpair[4]   = {9, 15, 15, 25};

// one thread per unnormalized wigner_3j element + fp64 atomic Frobenius norm
__global__ void e3_cg_elems(double* crUn, double* norms) {
    int i = blockIdx.x * blockDim.x + threadIdx.x;
    if (i >= 1156) return;
    int blk = 0;
    for (int t = 0; t < NBLK; ++t) if (i >= c_boff[t] && i < c_boff[t] + c_bsz[t]) blk = t;
    int e = i - c_boff[blk];
    int l1 = c_bl1[blk], l2 = c_bl2[blk], l3 = c_bl3[blk];
    int L1 = 2 * l1 + 1, L2 = 2 * l2 + 1, L3 = 2 * l3 + 1;
    int m = e % L3; int t2 = e / L3; int b = t2 % L2; int jj = t2 / L2;

    double q1r[25], q1i[25], q2r[25], q2i[25], q3r[81], q3i[81];
    dchange_basis(l1, q1r, q1i); dchange_basis(l2, q2r, q2i); dchange_basis(l3, q3r, q3i);

    // Cr[j,b,m] = Re( sum_{i,k,n} Q1[i,j] Q2[k,b] conj(Q3[n,m]) C[i,k,n] )
    double acc = 0.0;
    for (int ii = 0; ii < L1; ++ii)
        for (int k = 0; k < L2; ++k) {
            double r1 = q1r[ii * L1 + jj], i1 = q1i[ii * L1 + jj];
            double r2 = q2r[k * L2 + b],  i2 = q2i[k * L2 + b];
            double tr = r1 * r2 - i1 * i2, ti = r1 * i2 + i1 * r2;
            if (tr == 0.0 && ti == 0.0) continue;
            for (int nn = 0; nn < L3; ++nn) {
                double c = dsu2cg(l1, l2, l3, ii - l1, k - l2, nn - l3);
                if (c == 0.0) continue;
                double r3 = q3r[nn * L3 + m], i3 = q3i[nn * L3 + m];
                acc += (tr * r3 + ti * i3) * c;
            }
        }
    crUn[i] = acc;
    atomicAdd(&norms[blk], acc * acc);
}

// normalize per (pair,l3) block, scale by sqrt(2l3+1), write CG_T[r][ab]
__global__ void e3_cg_norm(const double* crUn, const double* norms, float* cg) {
    int i = blockIdx.x * blockDim.x + threadIdx.x;
    if (i >= 1156) return;
    int blk = 0;
    for (int t = 0; t < NBLK; ++t) if (i >= c_boff[t] && i < c_boff[t] + c_bsz[t]) blk = t;
    int e = i - c_boff[blk];
    int l2 = c_bl2[blk], l3 = c_bl3[blk];
    int L2 = 2 * l2 + 1, L3 = 2 * l3 + 1;
    int m = e % L3; int t2 = e / L3; int b = t2 % L2; int jj = t2 / L2;
    int pair = c_bpair[blk], R = c_Rpair[pair];
    double scale = sqrt((double)(2 * l3 + 1)) / sqrt(norms[blk]);
    cg[c_cgbase[pair] + (c_brofs[blk] + m) * R + jj * L2 + b] = (float)(crUn[i] * scale);
}

// ---------------------------------------------------------------------------
// Stage 1 worker: one 16x16 output tile of Hz = rme * CG_T for one pair type.
// Branch-free: A loads are unconditional from a clamped-safe address; K/N
// padding is handled by the zero-padded B fragments in LDS.
// ---------------------------------------------------------------------------
template <int R, int SH, int OFF, int KSTEPS, int NPAD, int CGPB>
__device__ __forceinline__ void gemm_pair(const float* __restrict__ X, int nEdge, int e0,
                                          const float* __restrict__ s_cgp,
                                          float* __restrict__ s_hz,
                                          int mtile, int ntile, int lh, int kh) {
    const int nch = 1 << SH;
    int row  = mtile * 16 + lh;                    // A row (edge,channel)
    int eloc = row >> SH, ch = row & (nch - 1);
    int ec   = imin2(e0 + eloc, nEdge - 1);        // clamp: address-safe, garbage ok
    const float* xr = X + (size_t)ec * 196 + OFF + ch * R;
    int n = ntile * 16 + lh;                       // B/D column (zero-padded in LDS)
    const float* bp = s_cgp + CGPB + ((kh * NPAD + n) << 1);

    v8f acc = {0.f, 0.f, 0.f, 0.f, 0.f, 0.f, 0.f, 0.f};
#pragma unroll
    for (int ks = 0; ks < KSTEPS; ++ks) {
        int k0 = ks * 4 + kh * 2;                  // A 16x4 f32: lane half holds K={k0,k0+1}
        v2f a, b;
        b.x = bp[ks * NPAD * 4 + 0];               // padded rows -> no conditionals
        b.y = bp[ks * NPAD * 4 + 1];
        if (ks * 4 + 3 < R) {                      // folds at unroll time
            a.x = xr[k0];
            a.y = xr[k0 + 1];
        } else {                                   // boundary step: value selects only
            int kc = imin2(k0, R - 2);
            float t0 = xr[kc], t1 = xr[kc + 1];
            float a0 = (k0 == kc) ? t0 : t1;
            a.x = (k0 < R)     ? a0 : 0.f;
            a.y = (k0 + 1 < R) ? t1 : 0.f;
        }
        acc = __builtin_amdgcn_wmma_f32_16x16x4_f32(false, a, false, b,
                                                    (short)0, acc, false, false);
    }
    if (n < R) {
#pragma unroll
        for (int g = 0; g < 8; ++g) {              // C/D: vgpr g -> M = g + 8*half
            int row2 = mtile * 16 + g + kh * 8;
            int e2 = row2 >> SH, c2 = row2 & (nch - 1);
            s_hz[e2 * 196 + OFF + n * nch + c2] = acc[g];
        }
    }
}

// ---------------------------------------------------------------------------
// Stage 2 helper: HR_q = DA * Hz_q * DB^T, Hz stored in LDS as [ab][q]
// ---------------------------------------------------------------------------
template <int L1, int L2, int NCH>
__device__ __forceinline__ void rotate_pair(const float* __restrict__ hzE, int off,
                                            const float* DA, const float* DB,
                                            float* __restrict__ outE) {
    for (int q = 0; q < NCH; ++q) {
        float H[L1][L2];
#pragma unroll
        for (int m = 0; m < L1; ++m)
#pragma unroll
            for (int o = 0; o < L2; ++o)
                H[m][o] = hzE[off + (m * L2 + o) * NCH + q];
        float T[L1][L2];
#pragma unroll
        for (int m = 0; m < L1; ++m)
#pragma unroll
            for (int k = 0; k < L2; ++k) {
                float s = 0.f;
#pragma unroll
                for (int o = 0; o < L2; ++o) s += H[m][o] * DB[k * L2 + o];
                T[m][k] = s;
            }
#pragma unroll
        for (int l = 0; l < L1; ++l)
#pragma unroll
            for (int k = 0; k < L2; ++k) {
                float s = 0.f;
#pragma unroll
                for (int m = 0; m < L1; ++m) s += DA[l * L1 + m] * T[m][k];
                outE[off + (q * L1 + l) * L2 + k] = s;
            }
    }
}

// ---------------------------------------------------------------------------
// Fused main kernel: 128 threads (4 wave32), 64 edges/block
// ---------------------------------------------------------------------------
__global__ __launch_bounds__(128) void e3_main(const float* __restrict__ X,
                                               const float* __restrict__ V,
                                               float* __restrict__ O,
                                               const float* __restrict__ CG,
                                               int nEdge) {
    __shared__ float s_cgp[CGP_TOT];               // zero-padded B fragments
    __shared__ float s_hz[EPB * 196];
    const int tid = threadIdx.x;
    const int e0 = blockIdx.x * EPB;

    // ---- build padded B fragments: [k/2][n][k&1], zero-padded K and N ------
    for (int i = tid; i < CGP_TOT; i += 128) s_cgp[i] = 0.f;
    __syncthreads();
    for (int i = tid; i < 1156; i += 128) {        // scatter dense CG_T -> padded
        int base, R, npad, loc;
        if (i < 81)       { base = CGP_PP; R = 9;  npad = 16; loc = i;       }
        else if (i < 306) { base = CGP_PD; R = 15; npad = 16; loc = i - 81;  }
        else if (i < 531) { base = CGP_DP; R = 15; npad = 16; loc = i - 306; }
        else              { base = CGP_DD; R = 25; npad = 32; loc = i - 531; }
        int k = loc / R, n = loc - k * R;
        s_cgp[base + (((k >> 1) * npad + n) << 1) + (k & 1)] = CG[i];
    }
    __syncthreads();

    // ---------------- Stage 1: WMMA GEMMs for p-p, p-d, d-p, d-d -----------
    const int wave = tid >> 5, lane = tid & 31;
    const int lh = lane & 15, kh = lane >> 4;
    // 40 tile jobs: 16 pp + 8 pd + 8 dp + 8 dd(4 Mtiles x 2 Ntiles); wave-uniform dispatch
    for (int j = wave; j < 40; j += 4) {
        if (j < 16)
            gemm_pair< 9, 2,  60, 3, 16, CGP_PP>(X, nEdge, e0, s_cgp, s_hz, j,      0,          lh, kh);
        else if (j < 24)
            gemm_pair<15, 1,  96, 4, 16, CGP_PD>(X, nEdge, e0, s_cgp, s_hz, j - 16, 0,          lh, kh);
        else if (j < 32)
            gemm_pair<15, 1, 141, 4, 16, CGP_DP>(X, nEdge, e0, s_cgp, s_hz, j - 24, 0,          lh, kh);
        else
            gemm_pair<25, 0, 171, 7, 32, CGP_DD>(X, nEdge, e0, s_cgp, s_hz, (j - 32) >> 1, (j - 32) & 1, lh, kh);
    }

    // identity-CG pairs: copy rme -> Hz (transposed to [r][q])
    for (int i = tid; i < EPB * 9; i += 128) {    // s-s off 0 R=1 q=9
        int e = i / 9, q = i - e * 9;
        if (e0 + e < nEdge) s_hz[e * 196 + q] = X[(size_t)(e0 + e) * 196 + q];
    }
    for (int i = tid; i < EPB * 18; i += 128) {   // s-p off 9 R=3 q=6
        int e = i / 18, t2 = i - e * 18, q = t2 / 3, r = t2 - q * 3;
        if (e0 + e < nEdge) s_hz[e * 196 + 9 + r * 6 + q] = X[(size_t)(e0 + e) * 196 + 9 + q * 3 + r];
    }
    for (int i = tid; i < EPB * 15; i += 128) {   // s-d off 27 R=5 q=3
        int e = i / 15, t2 = i - e * 15, q = t2 / 5, r = t2 - q * 5;
        if (e0 + e < nEdge) s_hz[e * 196 + 27 + r * 3 + q] = X[(size_t)(e0 + e) * 196 + 27 + q * 5 + r];
    }
    for (int i = tid; i < EPB * 18; i += 128) {   // p-s off 42 R=3 q=6
        int e = i / 18, t2 = i - e * 18, q = t2 / 3, r = t2 - q * 3;
        if (e0 + e < nEdge) s_hz[e * 196 + 42 + r * 6 + q] = X[(size_t)(e0 + e) * 196 + 42 + q * 3 + r];
    }
    for (int i = tid; i < EPB * 15; i += 128) {   // d-s off 126 R=5 q=3
        int e = i / 15, t2 = i - e * 15, q = t2 / 5, r = t2 - q * 5;
        if (e0 + e < nEdge) s_hz[e * 196 + 126 + r * 3 + q] = X[(size_t)(e0 + e) * 196 + 126 + q * 5 + r];
    }
    __syncthreads();

    // ---------------- Stage 2: per-edge Wigner rotations (VALU) ------------
    int eloc = tid >> 1, half = tid & 1;
    int edge = e0 + eloc;
    if (edge < nEdge) {
        float vx = V[(size_t)edge * 3 + 0], vy = V[(size_t)edge * 3 + 1], vz = V[(size_t)edge * 3 + 2];
        float inv = rsqrtf(vx * vx + vy * vy + vz * vz);
        vx = fminf(1.f, fmaxf(-1.f, vx * inv));
        vy = fminf(1.f, fmaxf(-1.f, vy * inv));
        vz = fminf(1.f, fmaxf(-1.f, vz * inv));
        float cb = vy, sb = sqrtf(fmaxf(0.f, 1.f - cb * cb));   // beta = acos(y)
        float h2 = vx * vx + vz * vz;                           // alpha = atan2(x, z)
        float ca, sa;
        if (h2 > 1e-30f) { float ih = rsqrtf(h2); ca = vz * ih; sa = vx * ih; }
        else             { ca = 1.f; sa = 0.f; }

        // D1 = Ry(alpha) @ Rx(beta) in e3nn (y,z,x) order
        float D1f[9] = { cb,      -sb,      0.f,
                         ca * sb,  ca * cb, -sa,
                         sa * sb,  sa * cb,  ca };
        // xyz coefficient rows of the rotation: x'=(a), y'=(b), z'=(c)
        float a1 = ca,  a2 = sa * sb, a3 = sa * cb;
        float b1 = 0.f, b2 = cb,      b3 = -sb;
        float c1 = -sa, c2 = ca * sb, c3 = ca * cb;
        const float S3 = 1.7320508075688772f, IS3 = 0.5773502691896258f, I2S3 = 0.28867513459481287f;
        float D2f[25];
        D2f[0]  = a1 * c3 + a3 * c1;  D2f[1]  = a1 * c2 + a2 * c1;
        D2f[2]  = (2.f * a2 * c2 - a1 * c1 - a3 * c3) * IS3;
        D2f[3]  = a2 * c3 + a3 * c2;  D2f[4]  = a3 * c3 - a1 * c1;
        D2f[5]  = a1 * b3 + a3 * b1;  D2f[6]  = a1 * b2 + a2 * b1;
        D2f[7]  = (2.f * a2 * b2 - a1 * b1 - a3 * b3) * IS3;
        D2f[8]  = a2 * b3 + a3 * b2;  D2f[9]  = a3 * b3 - a1 * b1;
        D2f[10] = S3 * b1 * b3;       D2f[11] = S3 * b1 * b2;
        D2f[12] = 0.5f * (2.f * b2 * b2 - b1 * b1 - b3 * b3);
        D2f[13] = S3 * b2 * b3;       D2f[14] = 0.5f * S3 * (b3 * b3 - b1 * b1);
        D2f[15] = b1 * c3 + b3 * c1;  D2f[16] = b1 * c2 + b2 * c1;
        D2f[17] = (2.f * b2 * c2 - b1 * c1 - b3 * c3) * IS3;
        D2f[18] = b2 * c3 + b3 * c2;  D2f[19] = b3 * c3 - b1 * c1;
        D2f[20] = c1 * c3 - a1 * a3;  D2f[21] = c1 * c2 - a1 * a2;
        D2f[22] = (2.f * (c2 * c2 - a2 * a2) - (c1 * c1 - a1 * a1) - (c3 * c3 - a3 * a3)) * I2S3;
        D2f[23] = c2 * c3 - a2 * a3;
        D2f[24] = 0.5f * ((c3 * c3 - a3 * a3) - (c1 * c1 - a1 * a1));

        const float D0f[1] = {1.f};
        const float* hzE = s_hz + eloc * 196;
        float* outE = O + (size_t)edge * 196;
        if (half == 0) {
            rotate_pair<1, 1, 9>(hzE,   0, D0f, D0f, outE);   // s-s
            rotate_pair<1, 3, 6>(hzE,   9, D0f, D1f, outE);   // s-p
            rotate_pair<1, 5, 3>(hzE,  27, D0f, D2f, outE);   // s-d
            rotate_pair<3, 3, 4>(hzE,  60, D1f, D1f, outE);   // p-p
            rotate_pair<5, 5, 1>(hzE, 171, D2f, D2f, outE);   // d-d
        } else {
            rotate_pair<3, 1, 6>(hzE,  42, D1f, D0f, outE);   // p-s
            rotate_pair<3, 5, 2>(hzE,  96, D1f, D2f, outE);   // p-d
            rotate_pair<5, 1, 3>(hzE, 126, D2f, D0f, outE);   // d-s
            rotate_pair<5, 3, 2>(hzE, 141, D2f, D1f, outE);   // d-p
        }
    }
}

// ---------------------------------------------------------------------------
extern "C" void kernel_launch(void* const* d_in, const int* in_sizes, int n_in,
                              void* d_out, int out_size, void* d_ws, size_t ws_size,
                              hipStream_t stream) {
    const float* X = (const float*)d_in[0];   // edge_features (n,196) f32
    const float* V = (const float*)d_in[1];   // edge_vectors  (n,3)   f32
    float* O = (float*)d_out;
    int n = in_sizes[1] / 3;

    // d_ws layout: [0,1156) f64 unnormalized Cr | [1156,1170) f64 norms |
    //              @16384B: 1156 f32 CG_T
    double* crUn  = (double*)d_ws;
    double* norms = crUn + 1156;
    float*  cg    = (float*)((char*)d_ws + 16384);

    hipMemsetAsync(d_ws, 0, (1156 + 14) * sizeof(double), stream);
    e3_cg_elems<<<(1156 + 127) / 128, 128, 0, stream>>>(crUn, norms);
    e3_cg_norm <<<(1156 + 127) / 128, 128, 0, stream>>>(crUn, norms, cg);

    int blocks = (n + EPB - 1) / EPB;
    if (blocks > 0)
        e3_main<<<blocks, 128, 0, stream>>>(X, V, O, cg, n);
}